// SparseRouter_77232101916871
// MI455X (gfx1250) — compile-verified
//
#include <hip/hip_runtime.h>
#include <hip/hip_bf16.h>

#define NUM_EXPERTS 16

typedef float v2f __attribute__((ext_vector_type(2)));
typedef float v4f __attribute__((ext_vector_type(4)));
typedef float v8f __attribute__((ext_vector_type(8)));

// ---------------------------------------------------------------------------
// Kernel 1: global average pool.  pooled[row] = mean(x[row*hw .. row*hw+hw-1])
// One wave32 per (b,c) row (rows are 4096 contiguous f32 = 16KB).
// Coalesced float4 loads with non-temporal hint (1 GiB streamed exactly once;
// don't pollute L2), then butterfly wave reduction.
// ---------------------------------------------------------------------------
__global__ void pool_mean_kernel(const float* __restrict__ x,
                                 float* __restrict__ pooled,
                                 int rows, int hw) {
  int wave = (int)((blockIdx.x * blockDim.x + threadIdx.x) >> 5);
  int lane = (int)(threadIdx.x & 31);
  if (wave >= rows) return;  // wave-uniform exit

  const v4f* p = (const v4f*)(x + (size_t)wave * (size_t)hw);
  int nv4 = hw >> 2;
  v4f acc = {0.f, 0.f, 0.f, 0.f};
  for (int i = lane; i < nv4; i += 32) {
    v4f v = __builtin_nontemporal_load(p + i);  // global_load_b128 ... th:NT
    acc += v;
  }
  float s = acc.x + acc.y + acc.z + acc.w;
#pragma unroll
  for (int off = 16; off > 0; off >>= 1)
    s += __shfl_xor(s, off, 32);
  if (lane == 0) pooled[wave] = s * (1.0f / (float)hw);
}

// ---------------------------------------------------------------------------
// Kernel 2: router. Single workgroup, 128 threads = 4 waves.
// GEMM logits[64,16] = pooled[64,C] @ gate_w.T[C,16] via V_WMMA_F32_16X16X4_F32.
// Wave w owns M-tile rows [16w, 16w+16); loops K in steps of 4.
//   A (16x4 f32): lanes 0-15 hold row M=lane, K={k0,k0+1}; lanes 16-31 K={k0+2,k0+3}
//   B (4x16 f32): lane n holds col N=n&15; lower half K={k0,k0+1}, upper K={k0+2,k0+3}
//   D (16x16 f32): VGPR j -> M = m0 + j (+8 for upper lane half), N = lane&15
// Then 64 threads each finish one batch row (softmax/top-2/EMA).
// ---------------------------------------------------------------------------
__global__ void router_kernel(const float* __restrict__ pooled,     // [64][C]
                              const float* __restrict__ gate_w,     // [16][C]
                              const float* __restrict__ gate_b,     // [16]
                              const float* __restrict__ expert_bias,// [16]
                              const float* __restrict__ usage_ema,  // [16]
                              float* __restrict__ out, int C) {
  __shared__ float s_logits[64 * NUM_EXPERTS];
  __shared__ float s_counts[NUM_EXPERTS];

  const int tid    = (int)threadIdx.x;
  const int waveId = tid >> 5;
  const int lane   = tid & 31;
  const int half   = lane >> 4;   // 0: K lo pair, 1: K hi pair
  const int lrow   = lane & 15;   // M index (A) / N index (B)
  const int m0     = waveId * 16;

  v8f acc = {0.f, 0.f, 0.f, 0.f, 0.f, 0.f, 0.f, 0.f};
  const float* aRow = pooled + (size_t)(m0 + lrow) * C;
  const float* bRow = gate_w + (size_t)lrow * C;

  for (int k0 = 0; k0 < C; k0 += 4) {
    int ka = k0 + 2 * half;
    v2f a, b;
    a.x = aRow[ka];
    a.y = aRow[ka + 1];
    b.x = bRow[ka];
    b.y = bRow[ka + 1];
    acc = __builtin_amdgcn_wmma_f32_16x16x4_f32(
        /*neg_a=*/false, a, /*neg_b=*/false, b,
        /*c_mod=*/(short)0, acc, /*reuse_a=*/false, /*reuse_b=*/false);
  }

#pragma unroll
  for (int j = 0; j < 8; ++j)
    s_logits[(m0 + 8 * half + j) * NUM_EXPERTS + lrow] = acc[j];

  if (tid < NUM_EXPERTS) s_counts[tid] = 0.f;
  __syncthreads();

  if (tid < 64) {
    float p[NUM_EXPERTS];
    float mx = -1e30f;
#pragma unroll
    for (int e = 0; e < NUM_EXPERTS; ++e) {
      float l = s_logits[tid * NUM_EXPERTS + e] + gate_b[e];
      l = fminf(fmaxf(l, -10.f), 10.f);   // clip logits
      l += expert_bias[e];
      p[e] = l;
      mx = fmaxf(mx, l);
    }
    float sum = 0.f;
#pragma unroll
    for (int e = 0; e < NUM_EXPERTS; ++e) { p[e] = expf(p[e] - mx); sum += p[e]; }
    float inv = 1.f / sum;
#pragma unroll
    for (int e = 0; e < NUM_EXPERTS; ++e)
      p[e] = fminf(fmaxf(p[e] * inv, 1e-6f), 1.f);

    // top-2 (ties -> lowest index, matches jax.lax.top_k)
    float m1 = -1e30f; int i1 = 0;
#pragma unroll
    for (int e = 0; e < NUM_EXPERTS; ++e)
      if (p[e] > m1) { m1 = p[e]; i1 = e; }
    float m2 = -1e30f; int i2 = 0;
#pragma unroll
    for (int e = 0; e < NUM_EXPERTS; ++e)
      if (e != i1 && p[e] > m2) { m2 = p[e]; i2 = e; }

    float denom = m1 + m2 + 1e-8f;
    out[tid * 2 + 0]       = m1 / denom;          // top_k_probs
    out[tid * 2 + 1]       = m2 / denom;
    out[128 + tid * 2 + 0] = (float)i1;           // top_k_indices
    out[128 + tid * 2 + 1] = (float)i2;
    atomicAdd(&s_counts[i1], 1.f);
    atomicAdd(&s_counts[i2], 1.f);
  }
  __syncthreads();

  if (tid == 0) out[256] = 0.f;                   // load_balance_loss
  if (tid < NUM_EXPERTS) {
    float usage = s_counts[tid] * (1.0f / (64.f * 2.f));
    float ema = 0.9f * usage_ema[tid] + 0.1f * usage;
    out[257 + tid] = ema;                         // new_usage_ema
    float imb = ema - 1.f / (float)NUM_EXPERTS;
    float nb = expert_bias[tid] - 0.01f * imb;
    out[273 + tid] = fminf(fmaxf(nb, -5.f), 5.f); // new_expert_bias
  }
}

extern "C" void kernel_launch(void* const* d_in, const int* in_sizes, int n_in,
                              void* d_out, int out_size, void* d_ws, size_t ws_size,
                              hipStream_t stream) {
  const float* x           = (const float*)d_in[0];
  const float* gate_w      = (const float*)d_in[1];
  const float* gate_b      = (const float*)d_in[2];
  const float* expert_bias = (const float*)d_in[3];
  const float* usage_ema   = (const float*)d_in[4];
  float* out = (float*)d_out;

  const int B  = 64;
  const int C  = in_sizes[1] / NUM_EXPERTS;                 // 1024
  const int hw = (int)((long long)in_sizes[0] / ((long long)B * C)); // 4096
  const int rows = B * C;                                   // 65536

  float* pooled = (float*)d_ws;                             // 256 KB scratch

  const int threads = 256;                    // 8 waves -> 8 rows per block
  const int rpb = threads / 32;
  const int blocks = (rows + rpb - 1) / rpb;
  pool_mean_kernel<<<blocks, threads, 0, stream>>>(x, pooled, rows, hw);
  router_kernel<<<1, 128, 0, stream>>>(pooled, gate_w, gate_b, expert_bias,
                                       usage_ema, out, C);
}